// GraphEncoder_19765439496837
// MI455X (gfx1250) — compile-verified
//
#include <hip/hip_runtime.h>

typedef __attribute__((ext_vector_type(2))) float v2f;
typedef __attribute__((ext_vector_type(8))) float v8f;

#define NN 20000     // nodes
#define NE 400000    // edges
#define BT 4096      // batch
#define IND 64       // in dim (layer 0)
#define HID 128      // hidden = HEADS * HEAD_D
#define HEAD_D 64

// ---------------------------------------------------------------------------
// Dual GEMM: xl = X @ Wl, xr = X @ Wr.  X:[NN x K] (row stride K), W:[K x 128].
// Register-blocked: one wave computes a 64x32 output block (4 M-tiles x 2
// N-tiles) of one matrix via 8 independent V_WMMA_F32_16X16X4_F32 per K-step.
// Wave decode: w = {m4, n2(0..3), mat(0..1)}; 8 waves/block share one m4.
// ---------------------------------------------------------------------------
#define MBLK 4
#define NBLK 2

__global__ __launch_bounds__(256)
void gemm_wmma_f32(const float* __restrict__ X, int K,
                   const float* __restrict__ Wl, const float* __restrict__ Wr,
                   float* __restrict__ xl, float* __restrict__ xr)
{
    const int lane = threadIdx.x & 31;
    const int w    = blockIdx.x * 8 + (threadIdx.x >> 5);
    const int mat  = w & 1;
    const int n2   = (w >> 1) & 3;
    const int m4   = w >> 3;
    const int rowBase = m4 * (MBLK * 16);
    if (rowBase >= NN) return;                    // wave-uniform: EXEC stays full

    const float* __restrict__ W = mat ? Wr : Wl;
    float* __restrict__ Y       = mat ? xr : xl;

    const int half = lane >> 4;                   // 0: K+0..1, 1: K+2..3
    const int l16  = lane & 15;
    const int colBase = n2 * (NBLK * 16);

    v8f c[MBLK][NBLK];
#pragma unroll
    for (int i = 0; i < MBLK; ++i)
#pragma unroll
        for (int j = 0; j < NBLK; ++j)
            c[i][j] = (v8f){0.f, 0.f, 0.f, 0.f, 0.f, 0.f, 0.f, 0.f};

    for (int kk = 0; kk < K; kk += 4) {
        const int k0 = kk + half * 2;
        v2f a[MBLK], b[NBLK];
#pragma unroll
        for (int i = 0; i < MBLK; ++i) {
            int r = rowBase + i * 16 + l16;
            r = r < NN ? r : NN - 1;              // clamp (values discarded on store)
            a[i].x = X[(size_t)r * K + k0];
            a[i].y = X[(size_t)r * K + k0 + 1];
        }
#pragma unroll
        for (int j = 0; j < NBLK; ++j) {
            const int col = colBase + j * 16 + l16;
            b[j].x = W[k0 * HID + col];
            b[j].y = W[(k0 + 1) * HID + col];
        }
#pragma unroll
        for (int i = 0; i < MBLK; ++i)
#pragma unroll
            for (int j = 0; j < NBLK; ++j)
                c[i][j] = __builtin_amdgcn_wmma_f32_16x16x4_f32(
                    false, a[i], false, b[j], (short)0, c[i][j], false, false);
    }

#pragma unroll
    for (int i = 0; i < MBLK; ++i) {
        const int rb = rowBase + i * 16 + half * 8;  // C: VGPR t -> M = t (+8 hi half)
#pragma unroll
        for (int j = 0; j < NBLK; ++j) {
            const int col = colBase + j * 16 + l16;
#pragma unroll
            for (int t = 0; t < 8; ++t) {
                const int r = rb + t;
                if (r < NN) Y[(size_t)r * HID + col] = c[i][j][t];
            }
        }
    }
}

// ---------------------------------------------------------------------------
// Edge score: e[E,H] = a_h . leaky_relu(xl[src] + xr[dst]); atomic segment max.
// Wave per edge; lanes 0-15 head0, 16-31 head1; 4 dims/lane.
// ---------------------------------------------------------------------------
__device__ __forceinline__ unsigned f32_key(float f) {
    unsigned u = __float_as_uint(f);
    return (u >> 31) ? ~u : (u | 0x80000000u);    // monotonic encoding
}

__global__ __launch_bounds__(256)
void edge_score(const int* __restrict__ ei, const float* __restrict__ xl,
                const float* __restrict__ xr, const float* __restrict__ aV,
                float* __restrict__ e, unsigned* __restrict__ m_enc)
{
    const int lane = threadIdx.x & 31;
    const int edge = blockIdx.x * 8 + (threadIdx.x >> 5);
    if (edge >= NE) return;
    const int src = ei[edge];
    const int dst = ei[NE + edge];
    const int h   = lane >> 4;
    const int d0  = h * HEAD_D + (lane & 15) * 4;

    const float4 l  = *(const float4*)(xl + (size_t)src * HID + d0);
    const float4 r  = *(const float4*)(xr + (size_t)dst * HID + d0);
    const float4 av = *(const float4*)(aV + d0);  // a is [H, 64] contiguous

    float s, p = 0.f;
    s = l.x + r.x; s = s > 0.f ? s : 0.2f * s; p += s * av.x;
    s = l.y + r.y; s = s > 0.f ? s : 0.2f * s; p += s * av.y;
    s = l.z + r.z; s = s > 0.f ? s : 0.2f * s; p += s * av.z;
    s = l.w + r.w; s = s > 0.f ? s : 0.2f * s; p += s * av.w;
#pragma unroll
    for (int off = 1; off < 16; off <<= 1)        // reduce within each 16-lane head
        p += __shfl_xor(p, off, 32);
    if ((lane & 15) == 0) {
        e[edge * 2 + h] = p;
        atomicMax(m_enc + dst * 2 + h, f32_key(p));
    }
}

__global__ void finalize_max(const unsigned* __restrict__ m_enc,
                             float* __restrict__ m_f)
{
    int i = blockIdx.x * blockDim.x + threadIdx.x;
    if (i >= NN * 2) return;
    unsigned key = m_enc[i];
    float f = (key & 0x80000000u) ? __uint_as_float(key ^ 0x80000000u)
                                  : __uint_as_float(~key);
    if (!(fabsf(f) < 3.0e38f)) f = 0.f;           // -inf (no edges) / NaN -> 0
    m_f[i] = f;
}

__global__ void edge_exp(const int* __restrict__ ei, const float* __restrict__ e,
                         const float* __restrict__ m_f, float* __restrict__ ex,
                         float* __restrict__ ssum)
{
    int id = blockIdx.x * blockDim.x + threadIdx.x;
    if (id >= NE * 2) return;
    int edge = id >> 1, h = id & 1;
    int dst = ei[NE + edge];
    float v = expf(e[id] - m_f[dst * 2 + h]);
    ex[id] = v;
    atomicAdd(ssum + dst * 2 + h, v);
}

// Weighted message scatter-add: acc[dst] += (ex/s)*ew * xl[src]. Wave per edge.
__global__ __launch_bounds__(256)
void edge_message(const int* __restrict__ ei, const float* __restrict__ ew,
                  const float* __restrict__ ex, const float* __restrict__ ssum,
                  const float* __restrict__ xl, float* __restrict__ acc)
{
    const int lane = threadIdx.x & 31;
    const int edge = blockIdx.x * 8 + (threadIdx.x >> 5);
    if (edge >= NE) return;
    const int src = ei[edge];
    const int dst = ei[NE + edge];
    const int h   = lane >> 4;
    const float coeff = ex[edge * 2 + h] / (ssum[dst * 2 + h] + 1e-16f) * ew[edge];
    const int d0 = h * HEAD_D + (lane & 15) * 4;
    const float4 l = *(const float4*)(xl + (size_t)src * HID + d0);
    float* o = acc + (size_t)dst * HID + d0;
    atomicAdd(o + 0, coeff * l.x);
    atomicAdd(o + 1, coeff * l.y);
    atomicAdd(o + 2, coeff * l.z);
    atomicAdd(o + 3, coeff * l.w);
}

__global__ void bias_elu(const float* __restrict__ acc, const float* __restrict__ b,
                         float* __restrict__ xout)
{
    int i = blockIdx.x * blockDim.x + threadIdx.x;
    if (i >= NN * HID) return;
    float v = acc[i] + b[i & (HID - 1)];
    xout[i] = v > 0.f ? v : expm1f(v);            // ELU (alpha=1)
}

__global__ void final_mean(const float* __restrict__ acc, const float* __restrict__ b,
                           float* __restrict__ xfin)
{
    int i = blockIdx.x * blockDim.x + threadIdx.x;
    if (i >= NN * HEAD_D) return;
    int node = i >> 6, c = i & 63;
    xfin[i] = 0.5f * (acc[(size_t)node * HID + c] + acc[(size_t)node * HID + 64 + c]) + b[c];
}

__global__ void gather_out(const int* __restrict__ pert, const float* __restrict__ xfin,
                           float* __restrict__ out)
{
    int i = blockIdx.x * blockDim.x + threadIdx.x;
    if (i >= BT * HEAD_D) return;
    int r = i >> 6, c = i & 63;
    out[i] = xfin[(size_t)pert[r] * HEAD_D + c];
}

// ---------------------------------------------------------------------------
extern "C" void kernel_launch(void* const* d_in, const int* in_sizes, int n_in,
                              void* d_out, int out_size, void* d_ws, size_t ws_size,
                              hipStream_t stream)
{
    const int*   pert = (const int*)d_in[0];
    const int*   ei   = (const int*)d_in[1];       // [2, NE]
    const float* ew   = (const float*)d_in[2];
    const float* emb  = (const float*)d_in[3];
    const float *Wl[4], *Wr[4], *aV[4], *bV[4];
    for (int i = 0; i < 4; ++i) {
        Wl[i] = (const float*)d_in[4 + i * 4 + 0];
        Wr[i] = (const float*)d_in[4 + i * 4 + 1];
        aV[i] = (const float*)d_in[4 + i * 4 + 2];
        bV[i] = (const float*)d_in[4 + i * 4 + 3];
    }

    float* ws = (float*)d_ws;
    float*    xbuf  = ws;                          // NN*HID  (layer in/out, final)
    float*    xl    = xbuf + (size_t)NN * HID;     // NN*HID
    float*    xr    = xl   + (size_t)NN * HID;     // NN*HID
    float*    acc   = xr   + (size_t)NN * HID;     // NN*HID
    float*    e     = acc  + (size_t)NN * HID;     // NE*2
    float*    ex    = e    + (size_t)NE * 2;       // NE*2
    unsigned* m_enc = (unsigned*)(ex + (size_t)NE * 2);   // NN*2
    float*    m_f   = (float*)(m_enc + (size_t)NN * 2);   // NN*2
    float*    ssum  = m_f + (size_t)NN * 2;               // NN*2

    // waves = Mblocks x 4 Nblocks x {Wl,Wr}; 8 waves per 256-thread block
    const int mBlocks = (NN + MBLK * 16 - 1) / (MBLK * 16);  // 313
    const int gemmBlocks = mBlocks;                           // 8 waves each

    for (int layer = 0; layer < 4; ++layer) {
        const float* X = (layer == 0) ? emb : xbuf;
        const int    K = (layer == 0) ? IND : HID;

        hipMemsetAsync(acc,   0, (size_t)NN * HID * sizeof(float), stream);
        hipMemsetAsync(m_enc, 0, (size_t)NN * 2 * sizeof(unsigned), stream);
        hipMemsetAsync(ssum,  0, (size_t)NN * 2 * sizeof(float), stream);

        gemm_wmma_f32<<<gemmBlocks, 256, 0, stream>>>(X, K, Wl[layer], Wr[layer], xl, xr);
        edge_score<<<(NE + 7) / 8, 256, 0, stream>>>(ei, xl, xr, aV[layer], e, m_enc);
        finalize_max<<<(NN * 2 + 255) / 256, 256, 0, stream>>>(m_enc, m_f);
        edge_exp<<<(NE * 2 + 255) / 256, 256, 0, stream>>>(ei, e, m_f, ex, ssum);
        edge_message<<<(NE + 7) / 8, 256, 0, stream>>>(ei, ew, ex, ssum, xl, acc);

        if (layer < 3)
            bias_elu<<<(NN * HID + 255) / 256, 256, 0, stream>>>(acc, bV[layer], xbuf);
        else
            final_mean<<<(NN * HEAD_D + 255) / 256, 256, 0, stream>>>(acc, bV[layer], xbuf);
    }

    gather_out<<<(BT * HEAD_D + 255) / 256, 256, 0, stream>>>(pert, xbuf, (float*)d_out);
}